// ADJ_FirstLayer_14740327759981
// MI455X (gfx1250) — compile-verified
//
#include <hip/hip_runtime.h>
#include <math.h>

// Output: adj (M x M), M = N + 4, fp32.
//   adj[i][i]   = 1.0   for i < N
//   adj[i][j]   = 0.25  for i >= N and j >= N   (bottom-right 4x4 block)
//   adj[i][j]   = 0.0   otherwise
//
// Pure store-bandwidth kernel: 268.7 MB of writes, ~0 FLOPs.
// 23.3 TB/s HBM => ~11.5 us floor. Strategy: one block per row,
// 128-bit non-temporal stores (output > 192 MB L2, never re-read on device).

typedef float float4v __attribute__((ext_vector_type(4)));

__global__ __launch_bounds__(256) void adj_fill_vec4(float* __restrict__ out,
                                                     int M, int N) {
    const int i      = blockIdx.x;   // row index (uniform -> SALU)
    const int q4     = M >> 2;       // float4 chunks per row (M % 4 == 0)
    const int diag4  = i >> 2;       // chunk containing the diagonal element
    const int lane   = i & 3;        // position of diagonal within its chunk
    const int blk4   = N >> 2;       // chunk holding columns N..N+3
    const bool bottom = (i >= N);    // last 4 rows

    float* __restrict__ rowp = out + (size_t)i * (size_t)M;

    for (int c = threadIdx.x; c < q4; c += 256) {
        float4v v = {0.0f, 0.0f, 0.0f, 0.0f};
        if (bottom) {
            if (c == blk4) {
                v = (float4v){0.25f, 0.25f, 0.25f, 0.25f};
            }
        } else if (c == diag4) {
            v[lane] = 1.0f;
        }
        // global_store_b128 with th:TH_STORE_NT — bypass-friendly: output
        // exceeds L2 and is only read back by the host.
        __builtin_nontemporal_store(v, (float4v*)(rowp + 4 * c));
    }
}

// Fallback for M not divisible by 4 (not hit for M = 8196, kept for safety).
__global__ __launch_bounds__(256) void adj_fill_scalar(float* __restrict__ out,
                                                       int M, int N) {
    const int i = blockIdx.x;
    const bool bottom = (i >= N);
    float* __restrict__ rowp = out + (size_t)i * (size_t)M;
    for (int j = threadIdx.x; j < M; j += 256) {
        float v = 0.0f;
        if (bottom) {
            if (j >= N) v = 0.25f;
        } else if (j == i) {
            v = 1.0f;
        }
        __builtin_nontemporal_store(v, rowp + j);
    }
}

extern "C" void kernel_launch(void* const* d_in, const int* in_sizes, int n_in,
                              void* d_out, int out_size, void* d_ws, size_t ws_size,
                              hipStream_t stream) {
    (void)d_in; (void)in_sizes; (void)n_in; (void)d_ws; (void)ws_size;
    if (out_size <= 0) return;

    // out_size = M*M; recover M via integer sqrt (M = 8196 for the reference).
    int M = (int)floor(sqrt((double)out_size) + 0.5);
    if ((long long)M * (long long)M != (long long)out_size) {
        // Defensive: search neighborhood for exact square root.
        for (int d = -2; d <= 2; ++d) {
            long long m = (long long)M + d;
            if (m > 0 && m * m == (long long)out_size) { M = (int)m; break; }
        }
    }
    const int N = M - 4;

    float* out = (float*)d_out;
    if ((M & 3) == 0) {
        adj_fill_vec4<<<M, 256, 0, stream>>>(out, M, N);
    } else {
        adj_fill_scalar<<<M, 256, 0, stream>>>(out, M, N);
    }
}